// Dual_Quantize5_43645457662418
// MI455X (gfx1250) — compile-verified
//
#include <hip/hip_runtime.h>

typedef float v2f __attribute__((ext_vector_type(2)));
typedef float v8f __attribute__((ext_vector_type(8)));

static constexpr int       kDim    = 256;
static constexpr int       kEmb    = 1024;
static constexpr long long kRows   = 32768;              // rows per input
static constexpr long long kElems  = kRows * kDim;       // 8388608 per input

// d_out layout (float element offsets), reference tuple order:
// ste_hr, ste_lr, ste_hr, ste_lr, diff x4, ind x4, dist x4
static constexpr long long STE0   = 0;
static constexpr long long SC0    = 4 * kElems;          // 33554432
static constexpr long long IND0   = SC0 + 4;             // 33554436
static constexpr long long DIST0  = IND0 + 4 * kRows;    // 33685508
static constexpr long long DISTSZ = kRows * (long long)kEmb; // 33554432

// workspace layout (float element offsets)
static constexpr long long WS_EMBT  = 0;        // 1024*256 transposed codebook
static constexpr long long WS_ENORM = 262144;   // 1024 column norms
static constexpr long long WS_PART  = 263168;   // 512 partial sums

// ---------------------------------------------------------------------------
// Kernel 0: transpose embed [256][1024] -> embT [1024][256], compute ||e_k||^2
// ---------------------------------------------------------------------------
__global__ __launch_bounds__(256) void vq_prep(const float* __restrict__ embed,
                                               float* __restrict__ ws) {
  __shared__ float part[16][16];
  const int k0  = (int)blockIdx.x * 16;
  const int kl  = (int)threadIdx.x >> 4;   // column within block
  const int seg = (int)threadIdx.x & 15;   // d-segment
  const int k   = k0 + kl;
  float acc = 0.f;
  for (int j = 0; j < 16; ++j) {
    const int d = seg * 16 + j;
    const float v = embed[(long long)d * kEmb + k];
    ws[WS_EMBT + (long long)k * kDim + d] = v;
    acc += v * v;
  }
  part[kl][seg] = acc;
  __syncthreads();
  if (seg == 0) {
    float s = 0.f;
    for (int j = 0; j < 16; ++j) s += part[kl][j];
    ws[WS_ENORM + k] = s;
  }
}

// ---------------------------------------------------------------------------
// Kernel 1: main — per block: 16 rows of one input, full K=1024 sweep.
// 8 waves x 8 subtiles of 16 cols; 64 chained v_wmma_f32_16x16x4_f32 each.
// ---------------------------------------------------------------------------
__global__ __launch_bounds__(256) void vq_main(const float* __restrict__ in_hr,
                                               const float* __restrict__ in_lr,
                                               const float* __restrict__ ws,
                                               float* __restrict__ out) {
  __shared__ float Atile[16 * 260];   // padded stride: conflict-free ds_load_b64
  __shared__ float xnorm[16];
  __shared__ float red_val[16 * 128];
  __shared__ int   red_idx[16 * 128];
  __shared__ int   sel[16];

  const int b   = (int)blockIdx.x;
  const int inp = b >> 11;           // 0 = hr, 1 = lr
  const int rt  = b & 2047;          // row tile
  const long long n0 = (long long)rt * 16;

  const float* __restrict__ in    = inp ? in_lr : in_hr;
  const float* __restrict__ embT  = ws + WS_EMBT;
  const float* __restrict__ enorm = ws + WS_ENORM;

  const long long ste_a  = STE0 + (long long)inp * kElems;
  const long long ste_b  = ste_a + 2 * kElems;
  const long long ind_a  = IND0 + (long long)inp * kRows;
  const long long ind_b  = ind_a + 2 * kRows;
  const long long dist_a = DIST0 + (long long)inp * DISTSZ;
  const long long dist_b = dist_a + 2 * DISTSZ;

  // ---- stage A tile (16 x 256) into LDS ----
  {
    const int r  = (int)threadIdx.x >> 4;
    const int c0 = ((int)threadIdx.x & 15) * 16;
    const float4* src = (const float4*)(in + (n0 + r) * kDim + c0);
    float4* dst = (float4*)(&Atile[r * 260 + c0]);
    for (int j = 0; j < 4; ++j) dst[j] = src[j];
  }
  __syncthreads();
  if (threadIdx.x < 16) {
    const float* row = &Atile[(int)threadIdx.x * 260];
    float s = 0.f;
    for (int j = 0; j < kDim; ++j) s += row[j] * row[j];
    xnorm[threadIdx.x] = s;
  }
  __syncthreads();

  const int wave  = (int)threadIdx.x >> 5;
  const int lane  = (int)threadIdx.x & 31;
  const int lhalf = lane >> 4;   // 0: K0/K1, rows 0-7 | 1: K2/K3, rows 8-15
  const int l16   = lane & 15;

  float xn[8];
  for (int v = 0; v < 8; ++v) xn[v] = xnorm[v + 8 * lhalf];

  float minv[8];
  int   mini[8];
  for (int v = 0; v < 8; ++v) { minv[v] = 3.4e38f; mini[v] = 0; }

  // A fragment source: 16x4 MxK layout — lane%16 = M row; lhalf picks K pair
  const float* arow = &Atile[l16 * 260 + 2 * lhalf];

  for (int s = 0; s < 8; ++s) {
    const int kc  = wave * 128 + s * 16;
    const int col = kc + l16;
    // B fragment source: 4x16 KxN — lane%16 = N col; lhalf picks K pair.
    // embT column is contiguous, so the 2 K-values are one b64 load.
    const float* bcol = embT + (long long)col * kDim + 2 * lhalf;
    const float  en   = enorm[col];
    __builtin_prefetch(bcol, 0, 0);

    v8f acc = {};
    #pragma unroll
    for (int d = 0; d < kDim; d += 4) {
      v2f a  = *(const v2f*)(arow + d);   // ds_load_b64
      v2f bb = *(const v2f*)(bcol + d);   // global_load_b64
      acc = __builtin_amdgcn_wmma_f32_16x16x4_f32(false, a, false, bb,
                                                  (short)0, acc, false, false);
    }

    // finalize dist = -2*(x.e) ... note acc holds x.e; dist uses -2*acc
    #pragma unroll
    for (int v = 0; v < 8; ++v) {
      const float dist = xn[v] - 2.0f * acc[v] + en;
      const long long nrow = n0 + v + 8 * lhalf;
      const long long o = nrow * kEmb + col;
      out[dist_a + o] = dist;
      out[dist_b + o] = dist;
      if (dist < minv[v]) { minv[v] = dist; mini[v] = col; }
    }
  }

  // ---- argmin reduction: 128 candidates per row ----
  for (int v = 0; v < 8; ++v) {
    const int row  = v + 8 * lhalf;
    const int slot = wave * 16 + l16;
    red_val[row * 128 + slot] = minv[v];
    red_idx[row * 128 + slot] = mini[v];
  }
  __syncthreads();

  if (threadIdx.x < 16) {
    const int row = (int)threadIdx.x;
    float bv = red_val[row * 128];
    int   bi = red_idx[row * 128];
    for (int j = 1; j < 128; ++j) {
      const float v = red_val[row * 128 + j];
      const int   i = red_idx[row * 128 + j];
      if (v < bv || (v == bv && i < bi)) { bv = v; bi = i; }
    }
    sel[row] = bi;
    out[ind_a + n0 + row] = (float)bi;
    out[ind_b + n0 + row] = (float)bi;
  }
  __syncthreads();

  // ---- gather quantized rows -> ste outputs (forward value of STE == q) ----
  {
    const int r  = (int)threadIdx.x >> 4;
    const int c0 = ((int)threadIdx.x & 15) * 16;
    const float4* q = (const float4*)(embT + (long long)sel[r] * kDim + c0);
    float4* da = (float4*)(out + ste_a + (n0 + r) * kDim + c0);
    float4* db = (float4*)(out + ste_b + (n0 + r) * kDim + c0);
    for (int j = 0; j < 4; ++j) { const float4 t = q[j]; da[j] = t; db[j] = t; }
  }
}

// ---------------------------------------------------------------------------
// Kernel 2: deterministic partial sums of (q - x)^2 per block.
// ---------------------------------------------------------------------------
__global__ __launch_bounds__(256) void vq_diff(const float* __restrict__ in_hr,
                                               const float* __restrict__ in_lr,
                                               const float* __restrict__ out,
                                               float* __restrict__ ws) {
  __shared__ float red[256];
  const int b   = (int)blockIdx.x;
  const int inp = b >> 8;
  const int blk = b & 255;
  const float* __restrict__ x = inp ? in_lr : in_hr;
  const float* __restrict__ q = out + (long long)inp * kElems;  // ste copy 1
  const long long base = (long long)blk * 32768 + threadIdx.x;
  float acc = 0.f;
  for (int j = 0; j < 128; ++j) {
    const long long idx = base + (long long)j * 256;
    const float d = q[idx] - x[idx];
    acc += d * d;
  }
  red[threadIdx.x] = acc;
  __syncthreads();
  for (int off = 128; off > 0; off >>= 1) {
    if ((int)threadIdx.x < off) red[threadIdx.x] += red[threadIdx.x + off];
    __syncthreads();
  }
  if (threadIdx.x == 0) ws[WS_PART + b] = red[0];
}

// ---------------------------------------------------------------------------
// Kernel 3: finish means, write 4 scalar slots.
// ---------------------------------------------------------------------------
__global__ __launch_bounds__(256) void vq_fin(const float* __restrict__ ws,
                                              float* __restrict__ out) {
  __shared__ float rh[256];
  __shared__ float rl[256];
  rh[threadIdx.x] = ws[WS_PART + threadIdx.x];
  rl[threadIdx.x] = ws[WS_PART + 256 + threadIdx.x];
  __syncthreads();
  for (int off = 128; off > 0; off >>= 1) {
    if ((int)threadIdx.x < off) {
      rh[threadIdx.x] += rh[threadIdx.x + off];
      rl[threadIdx.x] += rl[threadIdx.x + off];
    }
    __syncthreads();
  }
  if (threadIdx.x == 0) {
    const float dh = rh[0] / (float)kElems;
    const float dl = rl[0] / (float)kElems;
    out[SC0 + 0] = dh;
    out[SC0 + 1] = dl;
    out[SC0 + 2] = dh;
    out[SC0 + 3] = dl;
  }
}

extern "C" void kernel_launch(void* const* d_in, const int* in_sizes, int n_in,
                              void* d_out, int out_size, void* d_ws, size_t ws_size,
                              hipStream_t stream) {
  const float* in_hr = (const float*)d_in[0];
  const float* in_lr = (const float*)d_in[1];
  const float* embed = (const float*)d_in[2];
  float* out = (float*)d_out;
  float* ws  = (float*)d_ws;

  vq_prep<<<64, 256, 0, stream>>>(embed, ws);
  vq_main<<<4096, 256, 0, stream>>>(in_hr, in_lr, ws, out);
  vq_diff<<<512, 256, 0, stream>>>(in_hr, in_lr, out, ws);
  vq_fin<<<1, 256, 0, stream>>>(ws, out);
}